// Layer_16836271800329
// MI455X (gfx1250) — compile-verified
//
#include <hip/hip_runtime.h>

typedef __attribute__((ext_vector_type(2))) float v2f;
typedef __attribute__((ext_vector_type(8))) float v8f;

__device__ __forceinline__ v8f wmma4(v2f a, v2f b, v8f c) {
  // v_wmma_f32_16x16x4_f32 : D = A(16x4) * B(4x16) + C(16x16)
  return __builtin_amdgcn_wmma_f32_16x16x4_f32(false, a, false, b, (short)0, c, false, false);
}

// Async global->LDS copy, 16B per lane (GLOBAL_LOAD_ASYNC_TO_LDS_B128, ASYNCcnt).
// ldsaddr: byte offset in this wave's LDS allocation (per-lane VGPR)
// goff   : per-lane byte offset added to the SGPR base pair
__device__ __forceinline__ void async_ld_b128(unsigned ldsaddr, unsigned goff, const void* sbase) {
  asm volatile("global_load_async_to_lds_b128 %0, %1, %2 offset:0"
               :
               : "v"(ldsaddr), "v"(goff), "s"(sbase)
               : "memory");
}
__device__ __forceinline__ void wait_async0() {
  asm volatile("s_wait_asynccnt 0" ::: "memory");
}
__device__ __forceinline__ unsigned lds_off(const void* p) {
  // generic -> LDS addrspace cast, then ptrtoint = byte offset in LDS
  return (unsigned)(size_t)(__attribute__((address_space(3))) const char*)p;
}

__device__ __forceinline__ float2 cmul(float2 a, float2 b) {   // a*b
  return make_float2(fmaf(a.x, b.x, -a.y * b.y), fmaf(a.x, b.y, a.y * b.x));
}
__device__ __forceinline__ float2 cmulc(float2 a, float2 b) {  // a*conj(b)
  return make_float2(fmaf(a.x, b.x, a.y * b.y), fmaf(a.y, b.x, -a.x * b.y));
}
__device__ __forceinline__ float2 cadd(float2 a, float2 b) { return make_float2(a.x + b.x, a.y + b.y); }
__device__ __forceinline__ float2 cinv(float2 a) {
  float id = 1.0f / (a.x * a.x + a.y * a.y);
  return make_float2(a.x * id, -a.y * id);
}
__device__ __forceinline__ float2 crs(float2 a, float s) { return make_float2(a.x * s, a.y * s); }
__device__ __forceinline__ float wred(float v) {
#pragma unroll
  for (int o = 16; o; o >>= 1) v += __shfl_xor(v, o, 32);
  return v;
}

// One workgroup per batch sample b. 256 threads = 8 wave32; wave w owns user k=w
// for the small per-user stages; WMMA stages are tiled across waves.
__global__ void __launch_bounds__(256) wmmse_layer(
    const float2* __restrict__ Vg, const float2* __restrict__ Hg,
    const float* __restrict__ Xu, const float* __restrict__ Yu, const float* __restrict__ Zu,
    const float* __restrict__ Ou,
    const float* __restrict__ Xw, const float* __restrict__ Yw, const float* __restrict__ Zw,
    const float* __restrict__ Xv, const float* __restrict__ Yv, const float* __restrict__ Zv,
    const float* __restrict__ Ov,
    float2* __restrict__ outg) {
  // ---- manually overlaid LDS pool (62.5 KB) -------------------------------
  __shared__ __align__(16) float pool[15616];
  __shared__ float sTrV, sTrBre, sTrBim;

  float2* sH  = (float2*)(pool);           // [8][8][64] H            (phases 1-7)
  float2* sBm = (float2*)(pool);           // [64][64]   Bm           (phase 8, reuses H)
  float2* sV  = (float2*)(pool + 8192);    // [8][64][4] V            (phases 1-2)
  float2* sG  = (float2*)(pool + 8192);    // [8][64][4] G = H^H U W  (phases 6-8, reuses V)
  float2* sA  = (float2*)(pool + 12288);   // [8][8][8]  A            (phases 2-3)
  float2* sM  = (float2*)(pool + 12288);   // [8][8][8]  M = U W U^H  (phases 5-7, reuses A)
  float*  sPt = (pool + 12288);            // [3][64][8] P1/P2/P3     (phase 8, reuses A/M+HV)
  float2* sHV = (float2*)(pool + 13312);   // [8][8][4]  HV = H_k V_k (phases 2-4)
  float2* sQ  = (float2*)(pool + 13312);   // [8][64]    Q = M_k H_k  (phase 7, reuses HV+U)
  float2* sU  = (float2*)(pool + 13824);   // [8][8][4]  U            (phases 3-5)
  float2* sbp = (float2*)(pool + 13824);   // [64]       1/diag(Bm)   (phase 8, reuses U)
  float2* sW  = (float2*)(pool + 14336);   // [8][4][4]  W            (phases 4-5)
  float2* sUW = (float2*)(pool + 14592);   // [8][8][4]  UW = U W     (phases 5-6)
  float2* sCs = (float2*)(pool + 15104);   // [8][32]    per-wave scratch

  const int tid = threadIdx.x, lane = tid & 31, wave = tid >> 5;
  const int hi = lane >> 4, ll = lane & 15;
  const int b = blockIdx.x;
  const int k = wave;  // this wave's user index for wave-local stages

  if (tid == 0) { sTrV = 0.f; sTrBre = 0.f; sTrBim = 0.f; }

  // ---- phase 1: async DMA V,H into LDS (ASYNCcnt path) --------------------
  const float2* Vb = Vg + (size_t)b * 2048;  // [8][64][4], 16 KB
  const float2* Hb = Hg + (size_t)b * 4096;  // [8][8][64], 32 KB
  {
    const unsigned ldsH = lds_off(sH), ldsV = lds_off(sV);
#pragma unroll
    for (int i = 0; i < 8; i++) {  // H: 8 * 256 lanes * 16B = 32 KB
      unsigned off = (unsigned)(tid + i * 256) * 16u;
      async_ld_b128(ldsH + off, off, (const void*)Hb);
    }
#pragma unroll
    for (int i = 0; i < 4; i++) {  // V: 4 * 256 lanes * 16B = 16 KB
      unsigned off = (unsigned)(tid + i * 256) * 16u;
      async_ld_b128(ldsV + off, off, (const void*)Vb);
    }
    wait_async0();
  }

  // Warm L2 with the precoder weight matrices (global_prefetch_b8).
  if ((b & 255) == 0) {
    __builtin_prefetch((const char*)Xv + tid * 256, 0, 1);
    __builtin_prefetch((const char*)Yv + tid * 256, 0, 1);
    __builtin_prefetch((const char*)Zv + tid * 256, 0, 1);
  }
  __syncthreads();

  // trV = sum |V|^2 (read back from LDS after the async fill)
  float acc = 0.f;
#pragma unroll
  for (int i = 0; i < 8; i++) {
    float2 v = sV[tid + i * 256];
    acc = fmaf(v.x, v.x, fmaf(v.y, v.y, acc));
  }
  acc = wred(acc);
  if (lane == 0) atomicAdd(&sTrV, acc);
  __syncthreads();
  const float trV = sTrV;  // trV/P with P=1

  // ---- phase 2: C_kj = H_k V_j ; A_k = trV*I + sum_j C_kj C_kj^H ; HV=C_kk
  {
    const int r = lane >> 2, dc = lane & 3;       // C element (r,dc)
    const int e0 = lane * 2, e1 = e0 + 1;          // two A elements per lane
    const int i0 = e0 >> 3, j0 = e0 & 7, i1 = e1 >> 3, j1 = e1 & 7;
    float2 a0 = make_float2(0, 0), a1 = make_float2(0, 0);
    float2* cs = sCs + wave * 32;
    for (int j = 0; j < 8; j++) {
      float2 c = make_float2(0, 0);
#pragma unroll 8
      for (int n = 0; n < 64; n++)
        c = cadd(c, cmul(sH[(k * 8 + r) * 64 + n], sV[(j * 64 + n) * 4 + dc]));
      cs[r * 4 + dc] = c;
      if (j == k) sHV[(k * 8 + r) * 4 + dc] = c;
      __builtin_amdgcn_wave_barrier();
      float2 s0 = make_float2(0, 0), s1 = make_float2(0, 0);
#pragma unroll
      for (int d = 0; d < 4; d++) {
        s0 = cadd(s0, cmulc(cs[i0 * 4 + d], cs[j0 * 4 + d]));
        s1 = cadd(s1, cmulc(cs[i1 * 4 + d], cs[j1 * 4 + d]));
      }
      a0 = cadd(a0, s0);
      a1 = cadd(a1, s1);
      __builtin_amdgcn_wave_barrier();
    }
    if (i0 == j0) a0.x += trV;
    if (i1 == j1) a1.x += trV;
    sA[(k * 8 + i0) * 8 + j0] = a0;
    sA[(k * 8 + i1) * 8 + j1] = a1;
    __builtin_amdgcn_wave_barrier();
  }

  // ---- phase 3: U = (plus(A)X_U + A Y_U + Z_U) HV + O_U -------------------
  {
    const int i = lane >> 2, dc = lane & 3;
    float2 invA = cinv(sA[(k * 8 + i) * 8 + i]);
    float2 u = make_float2(0, 0);
#pragma unroll
    for (int j = 0; j < 8; j++) {
      float2 t = crs(invA, Xu[(k * 8 + i) * 8 + j]);
#pragma unroll
      for (int m = 0; m < 8; m++) t = cadd(t, crs(sA[(k * 8 + i) * 8 + m], Yu[(k * 8 + m) * 8 + j]));
      t.x += Zu[(k * 8 + i) * 8 + j];
      u = cadd(u, cmul(t, sHV[(k * 8 + j) * 4 + dc]));
    }
    u.x += Ou[(k * 8 + i) * 4 + dc];
    sU[(k * 8 + i) * 4 + dc] = u;
    __builtin_amdgcn_wave_barrier();
  }

  // ---- phase 4: E = I - U^H HV ; W = plus(E)X_W + E Y_W + Z_W -------------
  {
    float2* Es = sCs + wave * 32;
    if (lane < 16) {
      int i = lane >> 2, j = lane & 3;
      float2 e = make_float2(i == j ? 1.f : 0.f, 0.f);
#pragma unroll
      for (int r = 0; r < 8; r++)
        e = cadd(e, crs(cmulc(sHV[(k * 8 + r) * 4 + j], sU[(k * 8 + r) * 4 + i]), -1.f));
      Es[lane] = e;
    }
    __builtin_amdgcn_wave_barrier();
    if (lane < 16) {
      int i = lane >> 2, j = lane & 3;
      float2 invE = cinv(Es[i * 4 + i]);
      float2 w = crs(invE, Xw[(k * 4 + i) * 4 + j]);
#pragma unroll
      for (int m = 0; m < 4; m++) w = cadd(w, crs(Es[i * 4 + m], Yw[(k * 4 + m) * 4 + j]));
      w.x += Zw[(k * 4 + i) * 4 + j];
      sW[(k * 4 + i) * 4 + j] = w;
    }
    __builtin_amdgcn_wave_barrier();
  }

  // ---- phase 5: UW = U W ; trB += tr(UW U^H) ; M = UW U^H -----------------
  {
    const int r = lane >> 2, dc = lane & 3;
    float2 u = sU[(k * 8 + r) * 4 + dc];
    float2 uw = make_float2(0, 0);
#pragma unroll
    for (int e = 0; e < 4; e++) uw = cadd(uw, cmul(sU[(k * 8 + r) * 4 + e], sW[(k * 4 + e) * 4 + dc]));
    sUW[(k * 8 + r) * 4 + dc] = uw;
    float2 tb = cmulc(uw, u);
    float tre = wred(tb.x), tim = wred(tb.y);
    if (lane == 0) { atomicAdd(&sTrBre, tre); atomicAdd(&sTrBim, tim); }
    __builtin_amdgcn_wave_barrier();
#pragma unroll
    for (int e2 = 0; e2 < 2; e2++) {
      int e = lane * 2 + e2, rr = e >> 3, s = e & 7;
      float2 m = make_float2(0, 0);
#pragma unroll
      for (int d = 0; d < 4; d++) m = cadd(m, cmulc(sUW[(k * 8 + rr) * 4 + d], sU[(k * 8 + s) * 4 + d]));
      sM[(k * 8 + rr) * 8 + s] = m;
    }
  }
  __syncthreads();  // all waves done reading V -> safe to write G there

  // ---- phase 6: G = H^H UW ------------------------------------------------
#pragma unroll
  for (int e = 0; e < 8; e++) {
    int idx = e * 32 + lane;
    int n = idx >> 2, dc = idx & 3;
    float2 g = make_float2(0, 0);
#pragma unroll
    for (int r = 0; r < 8; r++)
      g = cadd(g, cmulc(sUW[(k * 8 + r) * 4 + dc], sH[(k * 8 + r) * 64 + n]));
    sG[(k * 64 + n) * 4 + dc] = g;
  }
  __syncthreads();
  const float2 trB = make_float2(sTrBre, sTrBim);

  // ---- phase 7 (WMMA): Bm = trB*I + sum_k H_k^H (M_k H_k) -----------------
  // Bm tiles live in registers: wave owns tiles 2w, 2w+1 of the 4x4 tiling.
  v8f accRe[2], accIm[2];
  const v8f z8 = {0, 0, 0, 0, 0, 0, 0, 0};
  accRe[0] = z8; accRe[1] = z8; accIm[0] = z8; accIm[1] = z8;
  const int t0 = wave * 2, t1 = wave * 2 + 1;
  for (int kk = 0; kk < 8; kk++) {
    // Q = M_kk H_kk  (8x64), all threads cooperate
#pragma unroll
    for (int e = 0; e < 2; e++) {
      int idx = e * 256 + tid, r = idx >> 6, n = idx & 63;
      float2 q = make_float2(0, 0);
#pragma unroll
      for (int s = 0; s < 8; s++) q = cadd(q, cmul(sM[(kk * 8 + r) * 8 + s], sH[(kk * 8 + s) * 64 + n]));
      sQ[r * 64 + n] = q;
    }
    __syncthreads();
#pragma unroll
    for (int tt = 0; tt < 2; tt++) {
      int t = tt ? t1 : t0;
      int tm = t >> 2, tn = t & 3;
      int ml = tm * 16 + ll, nl = tn * 16 + ll;
#pragma unroll
      for (int ks = 0; ks < 2; ks++) {
        int kk0 = ks * 4 + hi * 2;
        float2 h0 = sH[(kk * 8 + kk0) * 64 + ml], h1 = sH[(kk * 8 + kk0 + 1) * 64 + ml];
        float2 q0 = sQ[kk0 * 64 + nl], q1 = sQ[(kk0 + 1) * 64 + nl];
        v2f aRe = {h0.x, h1.x}, aIm = {h0.y, h1.y};
        v2f bRe = {q0.x, q1.x}, bIm = {q0.y, q1.y}, bReN = {-q0.x, -q1.x};
        // H^H Q : Re += Hre^T Qre + Him^T Qim ; Im += Hre^T Qim - Him^T Qre
        accRe[tt] = wmma4(aRe, bRe, accRe[tt]);
        accRe[tt] = wmma4(aIm, bIm, accRe[tt]);
        accIm[tt] = wmma4(aRe, bIm, accIm[tt]);
        accIm[tt] = wmma4(aIm, bReN, accIm[tt]);
      }
    }
    __syncthreads();
  }
  // add trB to the diagonal, write Bm into the (now dead) H region
#pragma unroll
  for (int tt = 0; tt < 2; tt++) {
    int t = tt ? t1 : t0;
    int tm = t >> 2, tn = t & 3;
#pragma unroll
    for (int r = 0; r < 8; r++) {
      int mloc = r + 8 * hi;
      float dsel = (tm == tn && ll == mloc) ? 1.f : 0.f;
      accRe[tt][r] += dsel * trB.x;
      accIm[tt][r] += dsel * trB.y;
    }
  }
#pragma unroll
  for (int tt = 0; tt < 2; tt++) {
    int t = tt ? t1 : t0;
    int tm = t >> 2, tn = t & 3;
#pragma unroll
    for (int r = 0; r < 8; r++) {
      int m = tm * 16 + r + 8 * hi, n = tn * 16 + ll;
      sBm[m * 64 + n] = make_float2(accRe[tt][r], accIm[tt][r]);
    }
  }
  __syncthreads();
  if (tid < 64) sbp[tid] = cinv(sBm[tid * 64 + tid]);
  __syncthreads();

  // ---- phase 8 (WMMA): out_k = diag(bp)P1 + Bm P2 + P3 + O_V --------------
  // P{1,2,3} = {X,Y,Z}_V[k] @ [G_re | G_im]  (real A, complex B packed in N)
  for (int kk = 0; kk < 8; kk++) {
    for (int jid = wave; jid < 12; jid += 8) {
      int w3 = jid >> 2, mt = jid & 3;
      const float* Wp = (w3 == 0 ? Xv : (w3 == 1 ? Yv : Zv)) + kk * 4096;
      v8f c = z8;
      int ml = mt * 16 + ll;
      int col = ll & 3;
#pragma unroll
      for (int ks = 0; ks < 16; ks++) {
        int kk0 = ks * 4 + hi * 2;
        v2f a = {Wp[ml * 64 + kk0], Wp[ml * 64 + kk0 + 1]};
        float2 g0 = sG[(kk * 64 + kk0) * 4 + col];
        float2 g1 = sG[(kk * 64 + kk0 + 1) * 4 + col];
        float b0 = (ll < 4) ? g0.x : ((ll < 8) ? g0.y : 0.f);
        float b1 = (ll < 4) ? g1.x : ((ll < 8) ? g1.y : 0.f);
        v2f bb = {b0, b1};
        c = wmma4(a, bb, c);
      }
      if (ll < 8) {
#pragma unroll
        for (int r = 0; r < 8; r++) {
          int m = mt * 16 + r + 8 * hi;
          sPt[(w3 * 64 + m) * 8 + ll] = c[r];
        }
      }
    }
    __syncthreads();
    if (wave < 4) {
      int mt = wave;
      v8f Dre = z8, Dim = z8;
      int ml = mt * 16 + ll;
      int col = ll & 3;
#pragma unroll
      for (int ks = 0; ks < 16; ks++) {
        int kk0 = ks * 4 + hi * 2;
        float2 m0 = sBm[ml * 64 + kk0], m1 = sBm[ml * 64 + kk0 + 1];
        v2f aRe = {m0.x, m1.x}, aIm = {m0.y, m1.y};
        float in4 = (ll < 4) ? 1.f : 0.f;
        float bre0 = in4 * sPt[(64 + kk0) * 8 + col];
        float bre1 = in4 * sPt[(64 + kk0 + 1) * 8 + col];
        float bim0 = in4 * sPt[(64 + kk0) * 8 + col + 4];
        float bim1 = in4 * sPt[(64 + kk0 + 1) * 8 + col + 4];
        v2f bRe = {bre0, bre1}, bIm = {bim0, bim1}, bImN = {-bim0, -bim1};
        // Bm P2 : Re = BmRe P2re - BmIm P2im ; Im = BmRe P2im + BmIm P2re
        Dre = wmma4(aRe, bRe, Dre);
        Dre = wmma4(aIm, bImN, Dre);
        Dim = wmma4(aRe, bIm, Dim);
        Dim = wmma4(aIm, bRe, Dim);
      }
      if (ll < 4) {
#pragma unroll
        for (int r = 0; r < 8; r++) {
          int m = mt * 16 + r + 8 * hi;
          float2 bp = sbp[m];
          float p1r = sPt[(0 * 64 + m) * 8 + ll], p1i = sPt[(0 * 64 + m) * 8 + ll + 4];
          float p3r = sPt[(2 * 64 + m) * 8 + ll], p3i = sPt[(2 * 64 + m) * 8 + ll + 4];
          float ore = Dre[r] + bp.x * p1r - bp.y * p1i + p3r + Ov[(kk * 64 + m) * 4 + ll];
          float oim = Dim[r] + bp.x * p1i + bp.y * p1r + p3i;
          outg[(((size_t)b * 8 + kk) * 64 + m) * 4 + ll] = make_float2(ore, oim);
        }
      }
    }
    __syncthreads();
  }
}

extern "C" void kernel_launch(void* const* d_in, const int* in_sizes, int n_in,
                              void* d_out, int out_size, void* d_ws, size_t ws_size,
                              hipStream_t stream) {
  (void)n_in; (void)out_size; (void)d_ws; (void)ws_size;
  const float2* V = (const float2*)d_in[0];
  const float2* H = (const float2*)d_in[1];
  const float* Xu = (const float*)d_in[2];
  const float* Yu = (const float*)d_in[3];
  const float* Zu = (const float*)d_in[4];
  const float* Ou = (const float*)d_in[5];
  const float* Xw = (const float*)d_in[6];
  const float* Yw = (const float*)d_in[7];
  const float* Zw = (const float*)d_in[8];
  const float* Xv = (const float*)d_in[9];
  const float* Yv = (const float*)d_in[10];
  const float* Zv = (const float*)d_in[11];
  const float* Ov = (const float*)d_in[12];
  // V is [B, K=8, Nt=64, d=4] complex64 -> 2048 complex elements per sample
  int B = in_sizes[0] / 2048;
  dim3 grid(B), block(256);
  hipLaunchKernelGGL(wmmse_layer, grid, block, 0, stream,
                     V, H, Xu, Yu, Zu, Ou, Xw, Yw, Zw, Xv, Yv, Zv, Ov,
                     (float2*)d_out);
}